// LongformerSelfAttention_77464030150835
// MI455X (gfx1250) — compile-verified
//
#include <hip/hip_runtime.h>

// ---------------------------------------------------------------------------
// Longformer self-attention for MI455X (gfx1250, wave32)
//   - v_wmma_f32_16x16x4_f32 for QK^T, W*V and the final FC GEMM
//   - global_load_async_to_lds_b128 (ASYNCcnt) for all tile staging
// ---------------------------------------------------------------------------
typedef __attribute__((ext_vector_type(2))) float v2f;
typedef __attribute__((ext_vector_type(8))) float v8f;

#define EMBED  512
#define HEADS  8
#define HDIM   64
#define SEQ    2048
#define BATCH  2
#define HALFW  128
#define QTILES (SEQ / 16)

#define KVSTR  (HDIM + 12)   // 76 floats = 304B: rows 16B-aligned, LDS-conflict-free
#define BTSTR  20            // 80B rows: 16B-aligned, conflict-free

// 1/sqrt(512)
#define INV_SCALE 0.04419417382415922f

#if defined(__AMDGCN__) && __has_builtin(__builtin_amdgcn_global_load_async_to_lds_b128) && \
    __has_builtin(__builtin_amdgcn_s_wait_asynccnt)
#define USE_ASYNC_LDS 1
#else
#define USE_ASYNC_LDS 0
#endif

#if USE_ASYNC_LDS
// The builtin's params are v4i pointers in AS1 (global src) / AS3 (LDS dst).
// Build those types via typedefs so the address-space attribute binds to the
// pointee, and produce the pointers via integer casts (low 32 bits of a
// generic LDS pointer are the LDS byte offset, flat-aperture rule ISA 10.2).
typedef int v4i __attribute__((__vector_size__(16)));
typedef __attribute__((address_space(1))) v4i* as1_v4i;
typedef __attribute__((address_space(3))) v4i* as3_v4i;

__device__ __forceinline__ void ld_async_b128(void* lds, const void* gsrc) {
  __builtin_amdgcn_global_load_async_to_lds_b128(
      (as1_v4i)(unsigned long long)gsrc,
      (as3_v4i)(unsigned int)(unsigned long long)lds,
      /*imm offset=*/0, /*cpol=*/0);
}
__device__ __forceinline__ void async_wait0() {
  __builtin_amdgcn_s_wait_asynccnt(0);
}
#endif

__device__ __forceinline__ v8f wmma4(v2f a, v2f b, v8f c) {
  // D = A(16x4,f32) * B(4x16,f32) + C(16x16,f32)
  return __builtin_amdgcn_wmma_f32_16x16x4_f32(
      /*neg_a=*/false, a, /*neg_b=*/false, b,
      /*c_mod=*/(short)0, c, /*reuse_a=*/false, /*reuse_b=*/false);
}

// ---------------------------------------------------------------------------
// Kernel 1: per-(b,h) masked sum of V over all keys + mask count per batch.
// ---------------------------------------------------------------------------
__global__ void sumv_kernel(const float* __restrict__ values,
                            const int* __restrict__ mask,
                            float* __restrict__ sumv,
                            float* __restrict__ cnt) {
  int bh = blockIdx.x;
  int b = bh / HEADS, h = bh % HEADS;
  int d = threadIdx.x;  // 0..63
  const float* vb = values + (size_t)b * SEQ * EMBED + h * HDIM + d;
  const int* mb = mask + b * SEQ;
  float acc = 0.f;
  int mcount = 0;
  for (int k = 0; k < SEQ; ++k) {
    int m = mb[k];
    if (m) acc += vb[(size_t)k * EMBED];
    mcount += (m != 0);
  }
  sumv[bh * HDIM + d] = acc;
  if (h == 0 && d == 0) cnt[b] = (float)mcount;
}

// ---------------------------------------------------------------------------
// Kernel 2: global attention for query position 0 (reference applies NO mask
// on this row).  One 256-thread block per (b,h).
// ---------------------------------------------------------------------------
__global__ void global_attn_kernel(const float* __restrict__ values,
                                   const float* __restrict__ keys,
                                   const float* __restrict__ query,
                                   float* __restrict__ attn) {
  __shared__ float e[SEQ];
  __shared__ float qs[HDIM];
  __shared__ float red[256];
  int bh = blockIdx.x;
  int b = bh / HEADS, h = bh % HEADS;
  int tid = threadIdx.x;
  const size_t base = (size_t)b * SEQ * EMBED + h * HDIM;

  if (tid < HDIM) qs[tid] = query[base + tid];  // query position 0
  __syncthreads();

  for (int k = tid; k < SEQ; k += 256) {
    const float* kr = keys + base + (size_t)k * EMBED;
    float dot = 0.f;
#pragma unroll
    for (int d2 = 0; d2 < HDIM; ++d2) dot += qs[d2] * kr[d2];
    e[k] = dot * INV_SCALE;
  }
  __syncthreads();

  float m = -3.402823466e38f;
  for (int k = tid; k < SEQ; k += 256) m = fmaxf(m, e[k]);
  red[tid] = m;
  __syncthreads();
  for (int s2 = 128; s2 > 0; s2 >>= 1) {
    if (tid < s2) red[tid] = fmaxf(red[tid], red[tid + s2]);
    __syncthreads();
  }
  m = red[0];
  __syncthreads();

  float ps = 0.f;
  for (int k = tid; k < SEQ; k += 256) {
    float a = __expf(e[k] - m);
    e[k] = a;
    ps += a;
  }
  __syncthreads();
  red[tid] = ps;
  __syncthreads();
  for (int s2 = 128; s2 > 0; s2 >>= 1) {
    if (tid < s2) red[tid] += red[tid + s2];
    __syncthreads();
  }
  float denom = red[0];
  __syncthreads();

  if (tid < HDIM) {
    const float* vb = values + base + tid;
    float acc = 0.f;
    for (int k = 0; k < SEQ; ++k) acc += e[k] * vb[(size_t)k * EMBED];
    attn[base + tid] = acc / denom;  // position 0
  }
}

// ---------------------------------------------------------------------------
// Kernel 3: banded attention for query positions 1..SEQ-1.
// One wave32 per (b, h, 16-query tile).  WMMA f32 16x16x4 for QK^T and W*V;
// K/V tiles staged via global_load_async_to_lds_b128.  Out-of-band uniform
// softmax contribution is folded in via sumv/cnt.
// ---------------------------------------------------------------------------
__global__ void __launch_bounds__(32) window_attn_kernel(
    const float* __restrict__ values, const float* __restrict__ keys,
    const float* __restrict__ query, const int* __restrict__ mask,
    const float* __restrict__ sumv, const float* __restrict__ cnt,
    float* __restrict__ attn) {
  __shared__ __align__(16) float Kt[16][KVSTR];  // key tile   (row=key, col=dim)
  __shared__ __align__(16) float Vt[16][KVSTR];  // value tile (row=key, col=dim)
  __shared__ float Wt[16][17];                   // weight tile (row=query, col=key)

  const int t = blockIdx.x;  // query tile, covers positions 1+16t .. 16+16t
  const int h = blockIdx.y;
  const int b = blockIdx.z;
  const int lane = threadIdx.x;
  const int ln = lane & 15;
  const int half = lane >> 4;

  const size_t base = (size_t)b * SEQ * EMBED + h * HDIM;

  // --- preload Q fragments in WMMA A layout (A VGPR v, lane-half -> K=v+2*half)
  int p = 1 + 16 * t + ln;  // query position handled by this lane's row
  int pc = (p < SEQ) ? p : (SEQ - 1);
  const float* qrow = query + base + (size_t)pc * EMBED;
  v2f qa[16];
#pragma unroll
  for (int c = 0; c < 16; ++c) {
    int kk = 4 * c + 2 * half;
    qa[c].x = qrow[kk];
    qa[c].y = qrow[kk + 1];
  }

  v8f oacc[4];
#pragma unroll
  for (int nt = 0; nt < 4; ++nt) oacc[nt] = {};
  float ps[8];
#pragma unroll
  for (int r = 0; r < 8; ++r) ps[r] = 0.f;

  const int jt0 = (t - 8 < 0) ? 0 : t - 8;
  const int jt1 = (t + 8 > QTILES - 1) ? QTILES - 1 : t + 8;

  for (int jt = jt0; jt <= jt1; ++jt) {
    const int j0 = jt * 16;
    __syncthreads();  // previous tile fully consumed before overwrite

    // --- stage K/V tiles (16 rows x 64 floats each = 256x16B per tile)
#if USE_ASYNC_LDS
#pragma unroll
    for (int it = 0; it < 8; ++it) {
      int f4 = lane + 32 * it;  // 0..255
      int r = f4 >> 4;          // key row 0..15
      int c4 = f4 & 15;         // 16B column 0..15
      const float* g = keys + base + (size_t)(j0 + r) * EMBED + c4 * 4;
      const float* gv = values + base + (size_t)(j0 + r) * EMBED + c4 * 4;
      ld_async_b128(&Kt[r][c4 * 4], g);
      ld_async_b128(&Vt[r][c4 * 4], gv);
    }
    // issue the per-column mask load now so it overlaps the async DMA
    const int mj = mask[b * SEQ + j0 + ln];
    async_wait0();
#else
    const int mj = mask[b * SEQ + j0 + ln];
#pragma unroll
    for (int it = 0; it < 8; ++it) {
      int f4 = lane + 32 * it;
      int r = f4 >> 4;
      int c4 = f4 & 15;
      const float4 kv = *(const float4*)(keys + base + (size_t)(j0 + r) * EMBED + c4 * 4);
      const float4 vv = *(const float4*)(values + base + (size_t)(j0 + r) * EMBED + c4 * 4);
      Kt[r][c4 * 4 + 0] = kv.x; Kt[r][c4 * 4 + 1] = kv.y;
      Kt[r][c4 * 4 + 2] = kv.z; Kt[r][c4 * 4 + 3] = kv.w;
      Vt[r][c4 * 4 + 0] = vv.x; Vt[r][c4 * 4 + 1] = vv.y;
      Vt[r][c4 * 4 + 2] = vv.z; Vt[r][c4 * 4 + 3] = vv.w;
    }
#endif
    __syncthreads();

    // prefetch next key tile while computing this one
    if (jt < jt1) {
      __builtin_prefetch(keys + base + (size_t)(j0 + 16 + (lane >> 1)) * EMBED + (lane & 1) * 32);
      __builtin_prefetch(values + base + (size_t)(j0 + 16 + (lane >> 1)) * EMBED + (lane & 1) * 32);
    }

    // --- S(16q x 16k) = Q * K^T : 16 wmma over the 64-dim, 2 chains for ILP
    v8f s0 = {}, s1 = {};
#pragma unroll
    for (int c = 0; c < 16; c += 2) {
      int kk0 = 4 * c + 2 * half;
      int kk1 = 4 * (c + 1) + 2 * half;
      v2f bf0, bf1;
      bf0.x = Kt[ln][kk0];      // B[K][N]: N = key = ln, K = dim slice
      bf0.y = Kt[ln][kk0 + 1];
      bf1.x = Kt[ln][kk1];
      bf1.y = Kt[ln][kk1 + 1];
      s0 = wmma4(qa[c], bf0, s0);
      s1 = wmma4(qa[c + 1], bf1, s1);
    }
    v8f s = s0 + s1;

    // --- weights: w = exp(s/scale)-1 inside band & unmasked, else 0
    const int j = j0 + ln;
#pragma unroll
    for (int r = 0; r < 8; ++r) {
      int i = 16 * t + r + 8 * half;  // band center (query pos = i+1)
      float w = 0.f;
      if (mj && i <= SEQ - 2 && j >= i - HALFW && j <= i + HALFW)
        w = __expf(s[r] * INV_SCALE) - 1.0f;
      ps[r] += w;
      Wt[r + 8 * half][ln] = w;  // C-layout -> LDS for relayout to A-layout
    }
    __syncthreads();

    // --- O(16q x 64d) += W(16x16) * V(16x64) : 4 n-tiles x 4 K-frags
#pragma unroll
    for (int c = 0; c < 4; ++c) {
      int kk = 4 * c + 2 * half;
      v2f a;
      a.x = Wt[ln][kk];       // A[M][K]: M = query = ln, K = key slice
      a.y = Wt[ln][kk + 1];
#pragma unroll
      for (int nt = 0; nt < 4; ++nt) {
        v2f bf;
        bf.x = Vt[kk][nt * 16 + ln];
        bf.y = Vt[kk + 1][nt * 16 + ln];
        oacc[nt] = wmma4(a, bf, oacc[nt]);
      }
    }
  }

  // --- row sums: reduce over the 16 lanes of each half
#pragma unroll
  for (int r = 0; r < 8; ++r) {
    float v = ps[r];
    v += __shfl_xor(v, 1);
    v += __shfl_xor(v, 2);
    v += __shfl_xor(v, 4);
    v += __shfl_xor(v, 8);
    ps[r] = v;
  }

  const float cb = cnt[b];
  const float* sv = sumv + (b * HEADS + h) * HDIM;
#pragma unroll
  for (int nt = 0; nt < 4; ++nt) {
    float svn = sv[nt * 16 + ln];
#pragma unroll
    for (int r = 0; r < 8; ++r) {
      int q = r + 8 * half;
      int pp = 1 + 16 * t + q;
      if (pp < SEQ) {
        float denom = ps[r] + cb;
        attn[base + (size_t)pp * EMBED + nt * 16 + ln] =
            (oacc[nt][r] + svn) / denom;
      }
    }
  }
}

// ---------------------------------------------------------------------------
// Kernel 4: out(4096 x 512) = attn @ fc_w^T + fc_b via WMMA f32 16x16x4.
// 4 waves/block; fc_w K-chunk shared across waves in LDS (async-staged);
// each wave owns a 16x64 output tile.
// ---------------------------------------------------------------------------
__global__ void __launch_bounds__(128) fc_kernel(const float* __restrict__ attn,
                                                 const float* __restrict__ w,
                                                 const float* __restrict__ bias,
                                                 float* __restrict__ out) {
  __shared__ __align__(16) float Bt[64][BTSTR];      // Bt[n][kk] = fc_w[o0+n][k0+kk]
  __shared__ __align__(16) float At[4][16][BTSTR];   // per-wave attn chunk [row][kk]

  const int wave = threadIdx.x >> 5;
  const int lane = threadIdx.x & 31;
  const int ln = lane & 15;
  const int half = lane >> 4;
  const int row0 = (blockIdx.x * 4 + wave) * 16;
  const int o0 = blockIdx.y * 64;

  v8f acc[4];
#pragma unroll
  for (int nt = 0; nt < 4; ++nt) acc[nt] = {};

  for (int k0 = 0; k0 < EMBED; k0 += 16) {
    __syncthreads();
    // fc_w chunk: 64 rows x 16 floats = 256x16B over 128 threads
    // attn chunk per wave: 16 rows x 16 floats = 64x16B over 32 lanes
#if USE_ASYNC_LDS
#pragma unroll
    for (int it = 0; it < 2; ++it) {
      int f4 = threadIdx.x + 128 * it;  // 0..255
      int n = f4 >> 2;
      int c4 = f4 & 3;
      ld_async_b128(&Bt[n][c4 * 4], w + (size_t)(o0 + n) * EMBED + k0 + c4 * 4);
    }
#pragma unroll
    for (int it = 0; it < 2; ++it) {
      int f4 = lane + 32 * it;  // 0..63
      int r = f4 >> 2;
      int c4 = f4 & 3;
      ld_async_b128(&At[wave][r][c4 * 4],
                    attn + (size_t)(row0 + r) * EMBED + k0 + c4 * 4);
    }
    async_wait0();
#else
#pragma unroll
    for (int it = 0; it < 2; ++it) {
      int f4 = threadIdx.x + 128 * it;
      int n = f4 >> 2;
      int c4 = f4 & 3;
      float4 v = *(const float4*)(w + (size_t)(o0 + n) * EMBED + k0 + c4 * 4);
      Bt[n][c4 * 4 + 0] = v.x; Bt[n][c4 * 4 + 1] = v.y;
      Bt[n][c4 * 4 + 2] = v.z; Bt[n][c4 * 4 + 3] = v.w;
    }
#pragma unroll
    for (int it = 0; it < 2; ++it) {
      int f4 = lane + 32 * it;
      int r = f4 >> 2;
      int c4 = f4 & 3;
      float4 v = *(const float4*)(attn + (size_t)(row0 + r) * EMBED + k0 + c4 * 4);
      At[wave][r][c4 * 4 + 0] = v.x; At[wave][r][c4 * 4 + 1] = v.y;
      At[wave][r][c4 * 4 + 2] = v.z; At[wave][r][c4 * 4 + 3] = v.w;
    }
#endif
    __syncthreads();

    // prefetch next K-chunk while this one computes
    if (k0 + 16 < EMBED) {
      __builtin_prefetch(w + (size_t)(o0 + (threadIdx.x >> 1)) * EMBED + k0 + 16 + (threadIdx.x & 1) * 8);
      __builtin_prefetch(attn + (size_t)(row0 + (lane >> 1)) * EMBED + k0 + 16 + (lane & 1) * 8);
    }

#pragma unroll
    for (int c = 0; c < 4; ++c) {
      int kk = 4 * c + 2 * half;
      v2f a;
      a.x = At[wave][ln][kk];
      a.y = At[wave][ln][kk + 1];
#pragma unroll
      for (int nt = 0; nt < 4; ++nt) {
        v2f bf;
        bf.x = Bt[nt * 16 + ln][kk];       // B[K][N] = fc_w[o0+N][k0+K]
        bf.y = Bt[nt * 16 + ln][kk + 1];
        acc[nt] = wmma4(a, bf, acc[nt]);
      }
    }
  }

#pragma unroll
  for (int nt = 0; nt < 4; ++nt) {
    float bb = bias[o0 + nt * 16 + ln];
#pragma unroll
    for (int r = 0; r < 8; ++r) {
      int row = row0 + r + 8 * half;
      out[(size_t)row * EMBED + o0 + nt * 16 + ln] = acc[nt][r] + bb;
    }
  }
}

// ---------------------------------------------------------------------------
extern "C" void kernel_launch(void* const* d_in, const int* in_sizes, int n_in,
                              void* d_out, int out_size, void* d_ws,
                              size_t ws_size, hipStream_t stream) {
  const float* values = (const float*)d_in[0];
  const float* keys   = (const float*)d_in[1];
  const float* query  = (const float*)d_in[2];
  const int*   mask   = (const int*)d_in[3];
  const float* fc_w   = (const float*)d_in[4];
  const float* fc_b   = (const float*)d_in[5];
  float* out = (float*)d_out;

  // workspace layout (floats): attn[B*S*EMBED] | sumv[B*H*64] | cnt[B]
  float* attn = (float*)d_ws;
  float* sumv = attn + (size_t)BATCH * SEQ * EMBED;
  float* cnt  = sumv + BATCH * HEADS * HDIM;

  sumv_kernel<<<BATCH * HEADS, 64, 0, stream>>>(values, mask, sumv, cnt);
  global_attn_kernel<<<BATCH * HEADS, 256, 0, stream>>>(values, keys, query, attn);
  window_attn_kernel<<<dim3(QTILES, HEADS, BATCH), 32, 0, stream>>>(
      values, keys, query, mask, sumv, cnt, attn);
  fc_kernel<<<dim3(BATCH * SEQ / 64, EMBED / 64), 128, 0, stream>>>(
      attn, fc_w, fc_b, out);
}